// GraphAE_5626407158312
// MI455X (gfx1250) — compile-verified
//
#include <hip/hip_runtime.h>

// GraphAE on MI455X (gfx1250, wave32).
// Exact simplification: pseudo==0 => spline coeffs c[:,0]=1, others 0, so
// SplineConv(x) = mean_agg(x) @ W[0] + x @ Wroot + b.
// Pipeline: deg -> agg1 -> WMMA GEMM(64->256,relu) -> agg2 ->
//           WMMA GEMM(256->64) -> WMMA GEMM(64->256,relu) -> WMMA GEMM(256->64).
// All GEMMs use V_WMMA_F32_16X16X4_F32 (exact f32, workload is scatter-bound
// so low-precision WMMA buys nothing here).

#define N_NODES 50000
#define N_EDGES 800000
#define CH_IN   64
#define CH_HID  256
#define CH_LAT  64

typedef __attribute__((ext_vector_type(2))) float v2f;
typedef __attribute__((ext_vector_type(8))) float v8f;

__global__ void zero_f32(float* __restrict__ p, long long n) {
    long long i = (long long)blockIdx.x * blockDim.x + threadIdx.x;
    if (i < n) p[i] = 0.0f;
}

// One thread per (edge, channel). agg[dst*C + c] += feat[src*C + c].
// Consecutive threads hit consecutive channels of the same dst row ->
// coalesced global_atomic_add_f32 bursts that resolve in L2 (192 MB).
__global__ void edge_agg(const int* __restrict__ src, const int* __restrict__ dst,
                         const float* __restrict__ feat, float* __restrict__ agg,
                         float* __restrict__ deg, int nedges, int cshift) {
    long long t = (long long)blockIdx.x * blockDim.x + threadIdx.x;
    const int C = 1 << cshift;
    long long total = (long long)nedges << cshift;
    if (t >= total) return;
    int e = (int)(t >> cshift);
    int c = (int)(t & (C - 1));
    int s = src[e];
    int d = dst[e];
    atomicAdd(&agg[(size_t)d * C + c], feat[(size_t)s * C + c]);
    if (deg && c == 0) atomicAdd(&deg[d], 1.0f);
}

__global__ void make_inv_deg(float* __restrict__ deg, int n) {
    int i = blockIdx.x * blockDim.x + threadIdx.x;
    if (i < n) deg[i] = 1.0f / fmaxf(deg[i], 1.0f);
}

// Fused dual-GEMM: out = act( (scale1 .* A1) @ W1 [+ A2 @ W2] + bias ).
// One wave per 16x16 output tile; K-loop in steps of 4 feeding
// v_wmma_f32_16x16x4_f32. A/B fragment layout per CDNA5 ISA:
//   lanes 0-15 carry K = {k0, k0+1}, lanes 16-31 carry K = {k0+2, k0+3}.
__global__ __launch_bounds__(256)
void gemm_dual_wmma(const float* __restrict__ A1, const float* __restrict__ scale1,
                    const float* __restrict__ W1,
                    const float* __restrict__ A2, const float* __restrict__ W2,
                    const float* __restrict__ bias, float* __restrict__ out,
                    int nrows, int kin, int kout, int relu) {
    const int lane  = threadIdx.x & 31;
    const int gwave = (int)((blockIdx.x * blockDim.x + threadIdx.x) >> 5);
    const int mtiles = nrows >> 4;           // nrows % 16 == 0 (50000/16 = 3125)
    const int ntiles = kout  >> 4;
    if (gwave >= mtiles * ntiles) return;    // wave-uniform: EXEC all-ones below
    const int tm = gwave % mtiles;
    const int tn = gwave / mtiles;

    const int mrow  = tm * 16 + (lane & 15); // A row this lane sources
    const int ncol  = tn * 16 + (lane & 15); // B/D column this lane sources
    const int khalf = (lane >> 4) * 2;       // 0 for lanes 0-15, 2 for 16-31

    v8f acc = {};

    {   // pass 1: (scale1 .* A1) @ W1  — per-row scale folded into A fragment
        const float s = scale1 ? scale1[mrow] : 1.0f;
        const float* __restrict__ arow = A1 + (size_t)mrow * kin;
        for (int k0 = 0; k0 < kin; k0 += 4) {
            v2f a, b;
            a.x = arow[k0 + khalf] * s;                        // global_load_b64
            a.y = arow[k0 + khalf + 1] * s;
            b.x = W1[(size_t)(k0 + khalf)     * kout + ncol];  // W is tiny: L2/WGP$ resident
            b.y = W1[(size_t)(k0 + khalf + 1) * kout + ncol];
            acc = __builtin_amdgcn_wmma_f32_16x16x4_f32(
                false, a, false, b, (short)0, acc, false, false);
        }
    }
    if (A2) {  // pass 2: + A2 @ W2 (root weight), same accumulator
        const float* __restrict__ arow = A2 + (size_t)mrow * kin;
        for (int k0 = 0; k0 < kin; k0 += 4) {
            v2f a, b;
            a.x = arow[k0 + khalf];
            a.y = arow[k0 + khalf + 1];
            b.x = W2[(size_t)(k0 + khalf)     * kout + ncol];
            b.y = W2[(size_t)(k0 + khalf + 1) * kout + ncol];
            acc = __builtin_amdgcn_wmma_f32_16x16x4_f32(
                false, a, false, b, (short)0, acc, false, false);
        }
    }

    const float bv = bias[ncol];
    // D layout: VGPR v holds M = tm*16 + v (lanes 0-15) / tm*16 + v + 8 (lanes 16-31)
    const int mbase = tm * 16 + (lane >> 4) * 8;
#pragma unroll
    for (int v = 0; v < 8; ++v) {
        float r = acc[v] + bv;
        if (relu) r = fmaxf(r, 0.0f);
        out[(size_t)(mbase + v) * kout + ncol] = r;
    }
}

extern "C" void kernel_launch(void* const* d_in, const int* in_sizes, int n_in,
                              void* d_out, int out_size, void* d_ws, size_t ws_size,
                              hipStream_t stream) {
    const float* x          = (const float*)d_in[0];
    const int*   edge_index = (const int*)  d_in[1];
    const float* conv1_W    = (const float*)d_in[2];   // [8,64,256]; only slab k=0 used
    const float* conv1_root = (const float*)d_in[3];
    const float* conv1_b    = (const float*)d_in[4];
    const float* conv2_W    = (const float*)d_in[5];   // [8,256,64]; only slab k=0 used
    const float* conv2_root = (const float*)d_in[6];
    const float* conv2_b    = (const float*)d_in[7];
    const float* dec_w1     = (const float*)d_in[8];
    const float* dec_b1     = (const float*)d_in[9];
    const float* dec_w2     = (const float*)d_in[10];
    const float* dec_b2     = (const float*)d_in[11];

    const int* src = edge_index;            // edge_index[0,:]
    const int* dst = edge_index + N_EDGES;  // edge_index[1,:]

    // Workspace: inv_deg[N] | agg[N*256] | h[N*256]  (= 102.6 MB)
    float* invdeg = (float*)d_ws;
    float* agg    = invdeg + N_NODES;
    float* h      = agg + (size_t)N_NODES * CH_HID;
    float* z      = (float*)d_out;          // z [N,64] staged in d_out, overwritten later
    float* h2     = agg;                    // decoder hidden reuses agg after layer 2

    const int mtiles = N_NODES / 16;        // 3125

    // ---- layer 1: degree + mean-aggregate x, then WMMA GEMM 64->256 (+root, relu)
    {
        long long nz = (long long)N_NODES * (1 + CH_IN);   // inv_deg + agg1 region
        zero_f32<<<(int)((nz + 255) / 256), 256, 0, stream>>>(invdeg, nz);
    }
    {
        long long tot = (long long)N_EDGES << 6;           // E * 64
        edge_agg<<<(int)((tot + 255) / 256), 256, 0, stream>>>(
            src, dst, x, agg, invdeg, N_EDGES, 6);
    }
    make_inv_deg<<<(N_NODES + 255) / 256, 256, 0, stream>>>(invdeg, N_NODES);
    {
        int waves = mtiles * (CH_HID / 16);
        gemm_dual_wmma<<<(waves + 7) / 8, 256, 0, stream>>>(
            agg, invdeg, conv1_W, x, conv1_root, conv1_b, h,
            N_NODES, CH_IN, CH_HID, 1);
    }

    // ---- layer 2: mean-aggregate h, WMMA GEMM 256->64 (+root), z -> d_out
    {
        long long nz = (long long)N_NODES * CH_HID;
        zero_f32<<<(int)((nz + 255) / 256), 256, 0, stream>>>(agg, nz);
    }
    {
        long long tot = (long long)N_EDGES << 8;           // E * 256
        edge_agg<<<(int)((tot + 255) / 256), 256, 0, stream>>>(
            src, dst, h, agg, nullptr, N_EDGES, 8);
    }
    {
        int waves = mtiles * (CH_LAT / 16);
        gemm_dual_wmma<<<(waves + 7) / 8, 256, 0, stream>>>(
            agg, invdeg, conv2_W, h, conv2_root, conv2_b, z,
            N_NODES, CH_HID, CH_LAT, 0);
    }

    // ---- decoder: h2 = relu(z @ dec_w1 + b1); out = h2 @ dec_w2 + b2
    {
        int waves = mtiles * (CH_HID / 16);
        gemm_dual_wmma<<<(waves + 7) / 8, 256, 0, stream>>>(
            z, nullptr, dec_w1, nullptr, nullptr, dec_b1, h2,
            N_NODES, CH_LAT, CH_HID, 1);
    }
    {
        int waves = mtiles * (CH_LAT / 16);
        gemm_dual_wmma<<<(waves + 7) / 8, 256, 0, stream>>>(
            h2, nullptr, dec_w2, nullptr, nullptr, dec_b2, (float*)d_out,
            N_NODES, CH_HID, CH_LAT, 0);
    }
}